// SingleHyperTKAN_74337293959663
// MI455X (gfx1250) — compile-verified
//
#include <hip/hip_runtime.h>
#include <math.h>

// ---------------------------------------------------------------------------
// SingleHyperTKAN on MI455X (gfx1250): f32 WMMA 16x16x4 for the two dominant
// hypergraph GEMMs, double-buffered async global->LDS staging; everything
// else latency-bound wave32 kernels.
// ---------------------------------------------------------------------------

typedef float v2f __attribute__((ext_vector_type(2)));
typedef float v8f __attribute__((ext_vector_type(8)));
typedef int   v4i_ __attribute__((vector_size(16)));

#define BB    8
#define TT    24
#define NNODE 1000
#define EEDGE 1000
#define F_INC 16
#define C_RAW_ 8
#define DMC   32
#define HCC   32
#define TKC   32
#define PREDC 12
#define NCOL  (TT*DMC)   /* 768 */
#define LAMC  0.3f
#define EPSC  1e-8f

#define AS_G __attribute__((address_space(1)))
#define AS_L __attribute__((address_space(3)))

#if __has_builtin(__builtin_amdgcn_global_load_async_to_lds_b128)
#define GEMM_ASYNC 1
#endif

__device__ __forceinline__ void async_wait0(){
#if defined(GEMM_ASYNC)
#if __has_builtin(__builtin_amdgcn_s_wait_asynccnt)
  __builtin_amdgcn_s_wait_asynccnt(0);
#else
  asm volatile("s_wait_asynccnt 0" ::: "memory");
#endif
#endif
}

// 16-byte global -> LDS copy: async on toolchains that expose the builtin.
__device__ __forceinline__ void cp_b128(const float* g, float* l){
#if defined(GEMM_ASYNC)
  __builtin_amdgcn_global_load_async_to_lds_b128(
      (AS_G v4i_*)(AS_G void*)(void*)g,
      (AS_L v4i_*)(AS_L void*)l,
      0, 0);
#else
  *reinterpret_cast<float4*>(l) = *reinterpret_cast<const float4*>(g);
#endif
}

__device__ __forceinline__ float sigmoidf_(float x){ return 1.0f/(1.0f+expf(-x)); }
__device__ __forceinline__ float siluf_(float x){ return x/(1.0f+expf(-x)); }

// --- 1) temporal mean/std features of x_raw: feat (B,N,16) -----------------
__global__ void feat_kernel(const float* __restrict__ xr, float* __restrict__ feat){
  int gid = blockIdx.x*blockDim.x + threadIdx.x;
  if (gid >= BB*NNODE*C_RAW_) return;
  int c = gid % C_RAW_; int n = (gid / C_RAW_) % NNODE; int b = gid / (C_RAW_*NNODE);
  float s = 0.f, s2 = 0.f;
  for (int t = 0; t < TT; ++t){
    float v = xr[((size_t)(b*TT+t)*NNODE + n)*C_RAW_ + c];
    s += v; s2 += v*v;
  }
  float m = s/(float)TT;
  float var = fmaxf(s2/(float)TT - m*m, 0.f);
  feat[((size_t)b*NNODE+n)*16 + c]     = m;
  feat[((size_t)b*NNODE+n)*16 + 8 + c] = sqrtf(var);
}

// --- 2) transpose H -> Ht (E,N) and edge degrees De = max(colsum,1) --------
__global__ void htde_kernel(const float* __restrict__ H, float* __restrict__ Ht,
                            float* __restrict__ De){
  int e = blockIdx.x;
  __shared__ float red[128];
  float s = 0.f;
  for (int n = threadIdx.x; n < NNODE; n += 128){
    float v = H[(size_t)n*EEDGE + e];
    Ht[(size_t)e*NNODE + n] = v;
    s += v;
  }
  red[threadIdx.x] = s; __syncthreads();
  for (int o = 64; o; o >>= 1){
    if (threadIdx.x < o) red[threadIdx.x] += red[threadIdx.x+o];
    __syncthreads();
  }
  if (threadIdx.x == 0) De[e] = fmaxf(red[0], 1.0f);
}

// --- 3) per-edge mean cosine similarity (one wave per (b,e)) ---------------
__global__ void edgesim_kernel(const float* __restrict__ feat, const int* __restrict__ members,
                               const int* __restrict__ centers, const int* __restrict__ offsets,
                               float* __restrict__ msim){
  int be = blockIdx.x; int b = be / EEDGE; int e = be % EEDGE;
  int o0 = offsets[e], o1 = offsets[e+1]; int cnt = o1 - o0;
  int cidx = centers[e];
  const float* cf = feat + ((size_t)b*NNODE + cidx)*16;
  float creg[16]; float nc = 0.f;
  #pragma unroll
  for (int k = 0; k < 16; ++k){ creg[k] = cf[k]; nc += creg[k]*creg[k]; }
  float cn = sqrtf(nc);
  float acc = 0.f;
  for (int j = threadIdx.x; j < cnt; j += 32){
    int m = members[o0 + j];
    const float* mf = feat + ((size_t)b*NNODE + m)*16;
    float d = 0.f, nm = 0.f;
    #pragma unroll
    for (int k = 0; k < 16; ++k){ float v = mf[k]; d += v*creg[k]; nm += v*v; }
    float sim = d / (sqrtf(nm)*cn + EPSC);
    acc += fminf(fmaxf(sim, 0.f), 1.f);
  }
  for (int off = 16; off > 0; off >>= 1) acc += __shfl_down(acc, off, 32);
  if (threadIdx.x == 0) msim[(size_t)b*EEDGE + e] = acc / fmaxf((float)cnt, 1.0f);
}

// --- 4) per-batch min-max normalize -> Wd and wtil = Wd/De -----------------
__global__ void wd_kernel(const float* __restrict__ msim, const float* __restrict__ W,
                          const float* __restrict__ De, float* __restrict__ wd,
                          float* __restrict__ wtil){
  int b = blockIdx.x;
  __shared__ float smin[256], smax[256];
  float lmin = 3.0e38f, lmax = -3.0e38f;
  for (int e = threadIdx.x; e < EEDGE; e += 256){
    float v = msim[(size_t)b*EEDGE + e];
    lmin = fminf(lmin, v); lmax = fmaxf(lmax, v);
  }
  smin[threadIdx.x] = lmin; smax[threadIdx.x] = lmax; __syncthreads();
  for (int o = 128; o; o >>= 1){
    if (threadIdx.x < o){
      smin[threadIdx.x] = fminf(smin[threadIdx.x], smin[threadIdx.x+o]);
      smax[threadIdx.x] = fmaxf(smax[threadIdx.x], smax[threadIdx.x+o]);
    }
    __syncthreads();
  }
  float mn = smin[0], mx = smax[0];
  for (int e = threadIdx.x; e < EEDGE; e += 256){
    float v = msim[(size_t)b*EEDGE + e];
    float nrm = (v - mn) / (mx - mn + EPSC);
    float w = W[e]*(1.0f + LAMC*nrm);
    wd  [(size_t)b*EEDGE + e] = w;
    wtil[(size_t)b*EEDGE + e] = w / De[e];
  }
}

// --- 5) node degrees dvi = rsqrt(clip(H @ Wd, eps)) (one wave per (b,n)) ---
__global__ void dvi_kernel(const float* __restrict__ H, const float* __restrict__ wd,
                           float* __restrict__ dvi){
  int bn = blockIdx.x; int b = bn / NNODE; int n = bn % NNODE;
  const float* hr = H  + (size_t)n*EEDGE;
  const float* wr = wd + (size_t)b*EEDGE;
  float s = 0.f;
  for (int e = threadIdx.x; e < EEDGE; e += 32) s += hr[e]*wr[e];
  for (int off = 16; off > 0; off >>= 1) s += __shfl_down(s, off, 32);
  if (threadIdx.x == 0) dvi[(size_t)b*NNODE + n] = rsqrtf(fmaxf(s, EPSC));
}

// --- 6) fused in-proj + dvi scale + transpose: xs[b][n][t*DM+f] ------------
__global__ void proj_kernel(const float* __restrict__ x, const float* __restrict__ w,
                            const float* __restrict__ bias, const float* __restrict__ dvi,
                            float* __restrict__ xs){
  int bn = blockIdx.x; int b = bn / NNODE; int n = bn % NNODE;
  __shared__ float Xs[TT*F_INC];
  __shared__ float Ws[F_INC*DMC];
  __shared__ float Bsv[DMC];
  for (int i = threadIdx.x; i < F_INC*DMC; i += 256) Ws[i] = w[i];
  if (threadIdx.x < DMC) Bsv[threadIdx.x] = bias[threadIdx.x];
  for (int i = threadIdx.x; i < TT*F_INC; i += 256){
    int t = i / F_INC, c = i % F_INC;
    Xs[i] = x[((size_t)(b*TT+t)*NNODE + n)*F_INC + c];
  }
  __syncthreads();
  float dv = dvi[(size_t)b*NNODE + n];
  for (int o = threadIdx.x; o < NCOL; o += 256){
    int t = o / DMC, f = o % DMC;
    float s = Bsv[f];
    #pragma unroll
    for (int c = 0; c < F_INC; ++c) s += Xs[t*F_INC + c]*Ws[c*DMC + f];
    xs[(size_t)bn*NCOL + o] = dv * s;
  }
}

// --- 7) f32 WMMA GEMM: C[b] = rowscale * (A @ B[b]) ------------------------
//   A: M x K row-major (shared across batch), B: K x NCOL, C: M x NCOL.
//   grid (NCOL/64, ceil(M/64), BB), block 128 (4 waves, 16x64 strip / wave).
//   Double-buffered A(64x32)/B(32x64) chunks staged via async global->LDS.
__global__ void __launch_bounds__(128)
gemm_wmma_f32(const float* __restrict__ A, const float* __restrict__ Bg,
              float* __restrict__ Cg, const float* __restrict__ scale,
              int M, int K){
  const int b = blockIdx.z;
  const float* Bp = Bg + (size_t)b*K*NCOL;
  float*       Cp = Cg + (size_t)b*M*NCOL;
  const float* sp = scale + (size_t)b*M;

  const int tid   = threadIdx.x;
  const int wave  = tid >> 5;
  const int lane  = tid & 31;
  const int lh    = lane >> 4;       // half-wave select (K split)
  const int lm    = lane & 15;       // M row / N col within tile
  const int n0    = blockIdx.x * 64;
  const int mbase = blockIdx.y * 64;
  const int m0    = mbase + wave*16;

  // Padded so row strides stay 8/16B aligned and LDS banks are spread.
  __shared__ float As[2][64][36];    // A chunk: 64 rows x 32 k
  __shared__ float Bs[2][32][68];    // B chunk: 32 k   x 64 cols
  v8f acc[4] = {};

  auto stage = [&](int sb, int kc){
    const int klen = (K - kc < 32) ? (K - kc) : 32;   // multiple of 4
    // A chunk: rows clamped into range; clamped rows feed accumulator rows
    // that the epilogue never stores, so no zero-fill needed.
    for (int idx = tid; idx < 512; idx += 128){
      int r = idx >> 3; int c4 = (idx & 7) << 2;      // r:0..63, c4:0..28
      if (c4 < klen){
        int grow = mbase + r; if (grow > M-1) grow = M-1;
        cp_b128(A + (size_t)grow*K + kc + c4, &As[sb][r][c4]);
      }
    }
    // B chunk (coalesced 16B segments)
    for (int idx = tid; idx < 512; idx += 128){
      int r = idx >> 4; int c4 = (idx & 15) << 2;     // r:0..31, c4:0..60
      if (r < klen)
        cp_b128(Bp + (size_t)(kc + r)*NCOL + n0 + c4, &Bs[sb][r][c4]);
    }
  };

  stage(0, 0);
  async_wait0();
  __syncthreads();

  const int mloc = wave*16 + lm;     // row of this lane within the A chunk
  for (int kc = 0; kc < K; kc += 32){
    const int sb   = (kc >> 5) & 1;
    const int klen = (K - kc < 32) ? (K - kc) : 32;
    if (kc + 32 < K) stage(sb ^ 1, kc + 32);          // overlap with compute
    if (kc + 64 < K)                                   // gfx1250 global_prefetch_b8
      __builtin_prefetch(Bp + (size_t)(kc + 64)*NCOL + n0, 0, 1);

    #pragma unroll 4
    for (int k = 0; k < klen; k += 4){
      const int kb = k + (lh << 1);
      v2f af = *reinterpret_cast<const v2f*>(&As[sb][mloc][kb]);  // ds_load_b64
      #pragma unroll
      for (int j = 0; j < 4; ++j){
        v2f bf;
        bf[0] = Bs[sb][kb    ][j*16 + lm];
        bf[1] = Bs[sb][kb + 1][j*16 + lm];
        acc[j] = __builtin_amdgcn_wmma_f32_16x16x4_f32(
            /*neg_a=*/false, af, /*neg_b=*/false, bf,
            /*c_mod=*/(short)0, acc[j], /*reuse_a=*/false, /*reuse_b=*/false);
      }
    }
    async_wait0();
    __syncthreads();
  }

  // epilogue: C/D layout -> VGPR r is row m0 + r + 8*lh, col n0 + j*16 + lm
  #pragma unroll
  for (int r = 0; r < 8; ++r){
    int row = m0 + r + (lh << 3);
    if (row < M){
      float s = sp[row];
      #pragma unroll
      for (int j = 0; j < 4; ++j)
        Cp[(size_t)row*NCOL + n0 + j*16 + lm] = acc[j][r] * s;
    }
  }
}

// --- 8) in-place conv + SiLU on h1 rows ------------------------------------
__global__ void conv_silu_kernel(float* __restrict__ h1, const float* __restrict__ cw,
                                 const float* __restrict__ cb){
  __shared__ float Ws[DMC*HCC];
  __shared__ float Bs2[HCC];
  for (int i = threadIdx.x; i < DMC*HCC; i += 128) Ws[i] = cw[i];
  if (threadIdx.x < HCC) Bs2[threadIdx.x] = cb[threadIdx.x];
  __syncthreads();
  int gid = blockIdx.x*128 + threadIdx.x;
  if (gid >= BB*NNODE*TT) return;
  float* row = h1 + (size_t)(gid / TT)*NCOL + (size_t)(gid % TT)*DMC;
  float in[DMC];
  #pragma unroll
  for (int c = 0; c < DMC; ++c) in[c] = row[c];
  #pragma unroll
  for (int f = 0; f < HCC; ++f){
    float s = Bs2[f];
    #pragma unroll
    for (int c = 0; c < DMC; ++c) s += in[c]*Ws[c*HCC + f];
    row[f] = siluf_(s);
  }
}

// --- 9) LSTM, one thread per (b,n) sequence row ----------------------------
__global__ void lstm_kernel(const float* __restrict__ hseq, const float* __restrict__ wx,
                            const float* __restrict__ wh, const float* __restrict__ bz,
                            float* __restrict__ hout){
  __shared__ float WX[HCC*4*TKC];
  __shared__ float WH[TKC*4*TKC];
  __shared__ float BZ[4*TKC];
  for (int i = threadIdx.x; i < HCC*4*TKC; i += 128) WX[i] = wx[i];
  for (int i = threadIdx.x; i < TKC*4*TKC; i += 128) WH[i] = wh[i];
  for (int i = threadIdx.x; i < 4*TKC; i += 128) BZ[i] = bz[i];
  __syncthreads();
  int gid = blockIdx.x*128 + threadIdx.x;
  if (gid >= BB*NNODE) return;
  const float* seq = hseq + (size_t)gid*NCOL;
  float h[TKC], c[TKC], nh[TKC], xt[HCC];
  #pragma unroll
  for (int j = 0; j < TKC; ++j){ h[j] = 0.f; c[j] = 0.f; }
  for (int t = 0; t < TT; ++t){
    #pragma unroll
    for (int k = 0; k < HCC; ++k) xt[k] = seq[t*HCC + k];
    for (int j = 0; j < TKC; ++j){
      float zi = BZ[j], zf = BZ[TKC+j], zg = BZ[2*TKC+j], zo = BZ[3*TKC+j];
      #pragma unroll
      for (int k2 = 0; k2 < HCC; ++k2){
        float xv = xt[k2];
        const float* wr = WX + k2*4*TKC;
        zi += xv*wr[j]; zf += xv*wr[TKC+j]; zg += xv*wr[2*TKC+j]; zo += xv*wr[3*TKC+j];
      }
      #pragma unroll
      for (int k2 = 0; k2 < TKC; ++k2){
        float hv = h[k2];
        const float* wr = WH + k2*4*TKC;
        zi += hv*wr[j]; zf += hv*wr[TKC+j]; zg += hv*wr[2*TKC+j]; zo += hv*wr[3*TKC+j];
      }
      float cc = sigmoidf_(zf)*c[j] + sigmoidf_(zi)*tanhf(zg);
      c[j] = cc;
      nh[j] = sigmoidf_(zo)*tanhf(cc);
    }
    #pragma unroll
    for (int j = 0; j < TKC; ++j) h[j] = nh[j];
  }
  for (int j = 0; j < TKC; ++j) hout[(size_t)gid*TKC + j] = h[j];
}

// --- 10) head + transposed output write ------------------------------------
__global__ void head_kernel(const float* __restrict__ ht, const float* __restrict__ w1,
                            const float* __restrict__ b1, const float* __restrict__ w2,
                            const float* __restrict__ b2, float* __restrict__ out, int olen){
  __shared__ float W1[TKC*TKC], B1[TKC], W2[TKC*PREDC], B2[PREDC];
  for (int i = threadIdx.x; i < TKC*TKC; i += 128) W1[i] = w1[i];
  for (int i = threadIdx.x; i < TKC*PREDC; i += 128) W2[i] = w2[i];
  if (threadIdx.x < TKC) B1[threadIdx.x] = b1[threadIdx.x];
  if (threadIdx.x < PREDC) B2[threadIdx.x] = b2[threadIdx.x];
  __syncthreads();
  int gid = blockIdx.x*128 + threadIdx.x;
  if (gid >= BB*NNODE) return;
  int b = gid / NNODE, n = gid % NNODE;
  float hr[TKC], a[TKC];
  #pragma unroll
  for (int j = 0; j < TKC; ++j) hr[j] = ht[(size_t)gid*TKC + j];
  #pragma unroll
  for (int j = 0; j < TKC; ++j){
    float s = B1[j];
    #pragma unroll
    for (int k = 0; k < TKC; ++k) s += hr[k]*W1[k*TKC + j];
    a[j] = siluf_(s);
  }
  for (int p = 0; p < PREDC; ++p){
    if (p >= olen) break;
    float s = B2[p];
    #pragma unroll
    for (int k = 0; k < TKC; ++k) s += a[k]*W2[k*PREDC + p];
    out[((size_t)(b*olen) + p)*NNODE + n] = s;   // (B, olen, N, 1)
  }
}

// ---------------------------------------------------------------------------
extern "C" void kernel_launch(void* const* d_in, const int* in_sizes, int n_in,
                              void* d_out, int out_size, void* d_ws, size_t ws_size,
                              hipStream_t stream){
  (void)in_sizes; (void)n_in; (void)ws_size;
  const float* x   = (const float*)d_in[0];
  const float* xr  = (const float*)d_in[1];
  const float* H   = (const float*)d_in[2];
  const float* W   = (const float*)d_in[3];
  const int*   mem = (const int*)d_in[4];
  const int*   cen = (const int*)d_in[5];
  const int*   off = (const int*)d_in[6];
  const float* ipw = (const float*)d_in[7];
  const float* ipb = (const float*)d_in[8];
  const float* cw  = (const float*)d_in[9];
  const float* cb  = (const float*)d_in[10];
  const float* lwx = (const float*)d_in[11];
  const float* lwh = (const float*)d_in[12];
  const float* lb  = (const float*)d_in[13];
  const float* hw1 = (const float*)d_in[14];
  const float* hb1 = (const float*)d_in[15];
  const float* hw2 = (const float*)d_in[16];
  const float* hb2 = (const float*)d_in[17];
  float* out = (float*)d_out;
  int olen = out_size / (BB*NNODE);   // OUT=1 -> output_length

  float* ws   = (float*)d_ws;
  float* feat = ws; ws += (size_t)BB*NNODE*16;
  float* msim = ws; ws += (size_t)BB*EEDGE;
  float* wd   = ws; ws += (size_t)BB*EEDGE;
  float* wtil = ws; ws += (size_t)BB*EEDGE;
  float* dvi  = ws; ws += (size_t)BB*NNODE;
  float* De   = ws; ws += 1024;
  float* Ht   = ws; ws += (size_t)EEDGE*NNODE;
  float* xs   = ws; ws += (size_t)BB*NNODE*NCOL;
  float* y1   = ws; ws += (size_t)BB*EEDGE*NCOL;
  float* htl  = ws; ws += (size_t)BB*NNODE*TKC;
  float* h1   = xs;   // xs is dead after gemm1; reuse for h1

  feat_kernel   <<<(BB*NNODE*C_RAW_ + 255)/256, 256, 0, stream>>>(xr, feat);
  htde_kernel   <<<EEDGE, 128, 0, stream>>>(H, Ht, De);
  edgesim_kernel<<<BB*EEDGE, 32, 0, stream>>>(feat, mem, cen, off, msim);
  wd_kernel     <<<BB, 256, 0, stream>>>(msim, W, De, wd, wtil);
  dvi_kernel    <<<BB*NNODE, 32, 0, stream>>>(H, wd, dvi);
  proj_kernel   <<<BB*NNODE, 256, 0, stream>>>(x, ipw, ipb, dvi, xs);

  dim3 g1(NCOL/64, (EEDGE + 63)/64, BB);   // y1 = wtil ∘ (Ht @ xs)
  gemm_wmma_f32<<<g1, 128, 0, stream>>>(Ht, xs, y1, wtil, EEDGE, NNODE);
  dim3 g2(NCOL/64, (NNODE + 63)/64, BB);   // h1 = dvi ∘ (H @ y1)
  gemm_wmma_f32<<<g2, 128, 0, stream>>>(H, y1, h1, dvi, NNODE, EEDGE);

  conv_silu_kernel<<<(BB*NNODE*TT + 127)/128, 128, 0, stream>>>(h1, cw, cb);
  lstm_kernel     <<<(BB*NNODE + 127)/128, 128, 0, stream>>>(h1, lwx, lwh, lb, htl);
  head_kernel     <<<(BB*NNODE + 127)/128, 128, 0, stream>>>(htl, hw1, hb1, hw2, hb2, out, olen);
}